// VAELoss_65283502899639
// MI455X (gfx1250) — compile-verified
//
#include <hip/hip_runtime.h>
#include <hip/hip_bf16.h>

typedef float v2f __attribute__((ext_vector_type(2)));
typedef float v8f __attribute__((ext_vector_type(8)));

#define EPSQ 1e-12f

// Problem dims (reference setup_inputs): B=8, N=8192, P=2048, D=512
#define B_ 8
#define N_ 8192
#define P_ 2048
#define D_ 512

#define TILE 1024          // target points staged in LDS per outer iteration (16 KB as float4)
#define WG_THREADS 256     // 8 wave32s per workgroup
#define ROWS_PER_WG 128    // 8 waves * 16 query rows each

// ---------------------------------------------------------------------------
// Zero the 4 accumulators in workspace (runs every launch -> deterministic).
// ---------------------------------------------------------------------------
__global__ void vae_zero_sums(float* sums) {
    if (threadIdx.x < 4) sums[threadIdx.x] = 0.0f;
}

// ---------------------------------------------------------------------------
// For each query point q in Q[b, nq, 3], find min_t ||q - t||^2 over T[b, nt, 3],
// then atomically accumulate sum_q sqrt(max(min, eps)) into sums[outIdx].
// One wave owns a 16-row tile; cross terms q.t for a 16x16 tile come from a
// single V_WMMA_F32_16X16X4_F32 (K = x,y,z,0 in fp32 -> reference precision).
// Target points are staged in LDS as (x, y, z, |t|^2) so the hot loop loads a
// single b128 per lane and needs no norm recomputation.
// ---------------------------------------------------------------------------
__global__ __launch_bounds__(WG_THREADS)
void vae_min_dist_sum(const float* __restrict__ Q, const float* __restrict__ T,
                      int nq, int nt, float* __restrict__ sums, int outIdx) {
    __shared__ float4 lds4[TILE];

    const int tid  = threadIdx.x;
    const int lane = tid & 31;
    const int wave = tid >> 5;
    const int wgPerBatch = nq / ROWS_PER_WG;
    const int b       = blockIdx.x / wgPerBatch;
    const int rowBase = (blockIdx.x % wgPerBatch) * ROWS_PER_WG + wave * 16;
    const bool hi = (lane >= 16);
    const int  l15 = lane & 15;

    const float* __restrict__ qb = Q + (size_t)b * nq * 3;
    const float* __restrict__ tb = T + (size_t)b * nt * 3;

    // A fragment (16x4, MxK): lanes 0-15 hold (K0,K1)=(x,y) of row M=lane;
    // lanes 16-31 hold (K2,K3)=(z,0) of row M=lane-16.
    const float* qp = qb + (size_t)(rowBase + l15) * 3;
    v2f afrag;
    afrag.x = hi ? qp[2] : qp[0];
    afrag.y = hi ? 0.0f  : qp[1];

    // Row norms for this lane's 8 accumulator rows: M = v + 8*hi (C/D layout).
    float qn8[8];
#pragma unroll
    for (int v = 0; v < 8; ++v) {
        const float* r = qb + (size_t)(rowBase + v + (hi ? 8 : 0)) * 3;
        qn8[v] = r[0]*r[0] + r[1]*r[1] + r[2]*r[2];
    }

    // Running min of (|t|^2 - 2 q.t); |q|^2 added after the loop.
    float rowmin[8];
#pragma unroll
    for (int v = 0; v < 8; ++v) rowmin[v] = 3.4e38f;

    for (int t0 = 0; t0 < nt; t0 += TILE) {
        __syncthreads();
        const float* src = tb + (size_t)t0 * 3;
        // Cooperative stage: pack (x,y,z,|t|^2) once per workgroup; reused by
        // all 8 waves x 64 column tiles.
        for (int idx = tid; idx < TILE; idx += WG_THREADS) {
            const float* s = src + idx * 3;
            const float x = s[0], y = s[1], z = s[2];
            lds4[idx] = make_float4(x, y, z, x * x + y * y + z * z);
        }
        if (t0 + TILE < nt)
            __builtin_prefetch(src + TILE * 3 + tid, 0, 0);  // global_prefetch_b8
        __syncthreads();

        for (int ct = 0; ct < TILE / 16; ++ct) {
            // B fragment (4x16, KxN): lanes 0-15 hold (K0,K1)=(x,y) of col N=lane;
            // lanes 16-31 hold (K2,K3)=(z,0) of col N=lane-16.
            const float4 p = lds4[ct * 16 + l15];     // single ds_load_b128
            const float pn = p.w;                     // |t|^2 for col N = l15
            v2f bfrag;
            bfrag.x = hi ? p.z  : p.x;
            bfrag.y = hi ? 0.0f : p.y;

            v8f acc = {0.f, 0.f, 0.f, 0.f, 0.f, 0.f, 0.f, 0.f};
            // D = A x B : 16x16 tile of q.t dot products, fp32 MACs.
            acc = __builtin_amdgcn_wmma_f32_16x16x4_f32(
                false, afrag, false, bfrag, (short)0, acc, false, false);

#pragma unroll
            for (int v = 0; v < 8; ++v) {
                float t = __builtin_fmaf(-2.0f, acc[v], pn);
                rowmin[v] = fminf(rowmin[v], t);
            }
        }
    }

    // Min across the 16 lanes of each half (each half covers cols N=0..15 mod 16).
#pragma unroll
    for (int m = 8; m >= 1; m >>= 1) {
#pragma unroll
        for (int v = 0; v < 8; ++v)
            rowmin[v] = fminf(rowmin[v], __shfl_xor(rowmin[v], m, 32));
    }

    // Lanes 0 and 16 hold final minima for rows rowBase+0..7 / rowBase+8..15.
    if (l15 == 0) {
        float s = 0.0f;
#pragma unroll
        for (int v = 0; v < 8; ++v)
            s += sqrtf(fmaxf(rowmin[v] + qn8[v], EPSQ));
        atomicAdd(&sums[outIdx], s);
    }
}

// ---------------------------------------------------------------------------
// KL: sum over all elements of -0.5*(1 + logvar - mu^2 - exp(logvar)).
// ---------------------------------------------------------------------------
__global__ void vae_kl_kernel(const float* __restrict__ mu,
                              const float* __restrict__ logvar,
                              float* __restrict__ sums) {
    const int i = blockIdx.x * blockDim.x + threadIdx.x;
    float v = 0.0f;
    if (i < B_ * D_) {
        const float m = mu[i], lv = logvar[i];
        v = -0.5f * (1.0f + lv - m * m - __expf(lv));
    }
#pragma unroll
    for (int s = 16; s >= 1; s >>= 1) v += __shfl_xor(v, s, 32);
    if ((threadIdx.x & 31) == 0) atomicAdd(&sums[3], v);
}

// ---------------------------------------------------------------------------
// Combine: (total, cd, kl, fidel)
// ---------------------------------------------------------------------------
__global__ void vae_finalize(const float* __restrict__ sums, float* __restrict__ out) {
    const float inv_bn = 1.0f / (float)(B_ * N_);
    const float cd  = 0.5f * (sums[0] * inv_bn + sums[1] * inv_bn);
    const float kl  = sums[3] / (float)B_;
    const float fid = sums[2] / (float)(B_ * P_);
    const float total = 1.0f * cd + 0.01f * kl + 0.5f * fid;
    out[0] = total; out[1] = cd; out[2] = kl; out[3] = fid;
}

extern "C" void kernel_launch(void* const* d_in, const int* in_sizes, int n_in,
                              void* d_out, int out_size, void* d_ws, size_t ws_size,
                              hipStream_t stream) {
    (void)in_sizes; (void)n_in; (void)out_size; (void)ws_size;
    const float* pred    = (const float*)d_in[0];  // [B,N,3]
    const float* full    = (const float*)d_in[1];  // [B,N,3]
    const float* partial = (const float*)d_in[2];  // [B,P,3]
    const float* mu      = (const float*)d_in[3];  // [B,D]
    const float* logvar  = (const float*)d_in[4];  // [B,D]
    float* out  = (float*)d_out;
    float* sums = (float*)d_ws;                    // 4 floats of scratch

    vae_zero_sums<<<1, 32, 0, stream>>>(sums);

    // d_pf: pred -> full
    vae_min_dist_sum<<<B_ * (N_ / ROWS_PER_WG), WG_THREADS, 0, stream>>>(
        pred, full, N_, N_, sums, 0);
    // d_fp: full -> pred
    vae_min_dist_sum<<<B_ * (N_ / ROWS_PER_WG), WG_THREADS, 0, stream>>>(
        full, pred, N_, N_, sums, 1);
    // fidelity: partial -> pred
    vae_min_dist_sum<<<B_ * (P_ / ROWS_PER_WG), WG_THREADS, 0, stream>>>(
        partial, pred, P_, N_, sums, 2);

    vae_kl_kernel<<<(B_ * D_ + 255) / 256, 256, 0, stream>>>(mu, logvar, sums);
    vae_finalize<<<1, 1, 0, stream>>>(sums, out);
}